// qcircuit2_89807766159615
// MI455X (gfx1250) — compile-verified
//
#include <hip/hip_runtime.h>
#include <math.h>

// ---------------------------------------------------------------------------
// qcircuit expectation on MI455X (gfx1250):
//   out[b,q] = <T, S_q^T S_q> / max(||v_b||^2, 1e-24),  T = Re(U' M U) 16x16
// Gram matrices computed with V_WMMA_F32_16X16X4_F32 (fp32 WMMA, wave32).
// 128MB state < 192MB L2 -> the 17 q-passes re-read from L2, not HBM.
// ---------------------------------------------------------------------------

typedef __attribute__((ext_vector_type(2))) float v2f;
typedef __attribute__((ext_vector_type(8))) float v8f;

__device__ __forceinline__ float2 cmul(float2 a, float2 b) {
    return make_float2(fmaf(a.x, b.x, -a.y * b.y), fmaf(a.x, b.y, a.y * b.x));
}
__device__ __forceinline__ float2 cadd(float2 a, float2 b) {
    return make_float2(a.x + b.x, a.y + b.y);
}
__device__ __forceinline__ float2 crs(float r, float2 z) {  // real * complex
    return make_float2(r * z.x, r * z.y);
}

// ---------------------------------------------------------------------------
// Kernel A: build T = Re(U^dag (X^{x4}) U) from the 36 weights. One block.
// ---------------------------------------------------------------------------
__global__ __launch_bounds__(256) void build_temp_kernel(
    const float* __restrict__ w, float* __restrict__ Tout) {
    __shared__ float2 mats[3][4][2][2];   // per layer, per qubit 2x2 YZY
    __shared__ float2 Lm[3][256];         // kron'ed 16x16 layers
    __shared__ float2 Ua[256], Ub[256];
    __shared__ int    sig[16];            // ring-of-CNOT permutation

    const int t = threadIdx.x;
    if (t < 12) {
        int l = t >> 2, k = t & 3;
        float t0 = w[12 * l + 3 * k + 0];
        float t1 = w[12 * l + 3 * k + 1];
        float t2 = w[12 * l + 3 * k + 2];
        float c0 = cosf(0.5f * t0), s0 = sinf(0.5f * t0);
        float cr = cosf(0.5f * t1), si = sinf(0.5f * t1);
        float c2 = cosf(0.5f * t2), s2 = sinf(0.5f * t2);
        float2 e  = make_float2(cr, -si);   // exp(-i t1/2)
        float2 ec = make_float2(cr,  si);   // conj
        // M = Ry(t2) Rz(t1) Ry(t0)
        mats[l][k][0][0] = cadd(crs( c2 * c0, e), crs(-s2 * s0, ec));
        mats[l][k][0][1] = cadd(crs(-c2 * s0, e), crs(-s2 * c0, ec));
        mats[l][k][1][0] = cadd(crs( s2 * c0, e), crs( c2 * s0, ec));
        mats[l][k][1][1] = cadd(crs(-s2 * s0, e), crs( c2 * c0, ec));
    }
    if (t < 16) {
        // apply cnot(0,1), cnot(1,2), cnot(2,3), cnot(3,0); qubit k = bit (3-k)
        int x = t;
        x ^= ((x >> 3) & 1) << 2;
        x ^= ((x >> 2) & 1) << 1;
        x ^= ((x >> 1) & 1) << 0;
        x ^= (x & 1) << 3;
        sig[t] = x;
    }
    __syncthreads();

    const int r = t >> 4, c = t & 15;
    #pragma unroll
    for (int l = 0; l < 3; ++l) {
        float2 p = make_float2(1.f, 0.f);
        #pragma unroll
        for (int k = 0; k < 4; ++k)
            p = cmul(p, mats[l][k][(r >> (3 - k)) & 1][(c >> (3 - k)) & 1]);
        Lm[l][t] = p;
    }
    __syncthreads();

    // U = L0
    Ua[t] = Lm[0][t];
    __syncthreads();
    // U = ring @ U  (row scatter: row x -> row sig(x))
    Ub[sig[r] * 16 + c] = Ua[t];
    __syncthreads();
    // U = L1 @ U
    {
        float2 s = make_float2(0.f, 0.f);
        #pragma unroll
        for (int k = 0; k < 16; ++k) s = cadd(s, cmul(Lm[1][r * 16 + k], Ub[k * 16 + c]));
        Ua[t] = s;
    }
    __syncthreads();
    Ub[sig[r] * 16 + c] = Ua[t];
    __syncthreads();
    // U = L2 @ U   (final)
    {
        float2 s = make_float2(0.f, 0.f);
        #pragma unroll
        for (int k = 0; k < 16; ++k) s = cadd(s, cmul(Lm[2][r * 16 + k], Ub[k * 16 + c]));
        Ua[t] = s;
    }
    __syncthreads();
    // temp[i][j] = sum_k conj(U[k][i]) * U[15^k][j];  M = X^{x4} flips all bits.
    float acc = 0.f;
    #pragma unroll
    for (int k = 0; k < 16; ++k) {
        float2 a = Ua[k * 16 + r];
        float2 b = Ua[(15 - k) * 16 + c];
        acc = fmaf(a.x, b.x, fmaf(a.y, b.y, acc));   // Re(conj(a)*b)
    }
    Tout[t] = acc;
}

// ---------------------------------------------------------------------------
// Kernel B: per-batch sum of squares (also warms the 192MB L2 with the state)
// ---------------------------------------------------------------------------
__global__ __launch_bounds__(256) void sumsq_kernel(
    const float* __restrict__ vec, float* __restrict__ sumsq) {
    const int b = blockIdx.x, tid = threadIdx.x;
    const float4* v4 = (const float4*)(vec + ((size_t)b << 20));
    float acc = 0.f;
    for (int j = tid; j < (1 << 18); j += 256) {
        float4 x = v4[j];
        acc = fmaf(x.x, x.x, acc);
        acc = fmaf(x.y, x.y, acc);
        acc = fmaf(x.z, x.z, acc);
        acc = fmaf(x.w, x.w, acc);
    }
    #pragma unroll
    for (int o = 16; o; o >>= 1) acc += __shfl_xor(acc, o, 32);
    __shared__ float wsum[8];
    if ((tid & 31) == 0) wsum[tid >> 5] = acc;
    __syncthreads();
    if (tid == 0) {
        float s = 0.f;
        #pragma unroll
        for (int i = 0; i < 8; ++i) s += wsum[i];
        sumsq[b] = s;
    }
}

// ---------------------------------------------------------------------------
// Kernel C: Gram quadratic forms via V_WMMA_F32_16X16X4_F32.
// grid = (32 K-chunks, 17 q, 32 batch), 256 threads (8 wave32).
// For axis position q: element (row r, col i) lives at
//   (a << (shift+4)) + (i << shift) + c,  a=r>>shift, c=r&((1<<shift)-1), shift=16-q.
// Each block: 8 tiles of 256 rows; per tile each wave runs 8 WMMA K=4 steps.
// ---------------------------------------------------------------------------
__global__ __launch_bounds__(256) void gram_kernel(
    const float* __restrict__ vec, const float* __restrict__ Tg,
    float* __restrict__ raw) {
    __shared__ float tile[256][17];   // 17-float pitch: conflict-free banks
    __shared__ float Tl[256];

    const int tid   = threadIdx.x;
    const int q     = blockIdx.y;
    const int b     = blockIdx.z;
    const int shift = 16 - q;

    Tl[tid] = Tg[tid];
    const float* vb = vec + ((size_t)b << 20);

    const int lane = tid & 31;
    const int wv   = tid >> 5;
    const int h    = lane >> 4;   // K half-select (A: K=2h,2h+1; B: K=h,2+h)
    const int m    = lane & 15;   // column within 16

    v8f acc = {};   // 16x16 f32 accumulator (8 VGPRs/lane)

    for (int t = 0; t < 8; ++t) {
        const unsigned r    = (blockIdx.x * 8u + (unsigned)t) * 256u + (unsigned)tid;
        const unsigned a    = r >> shift;
        const unsigned c    = r & ((1u << shift) - 1u);
        const unsigned base = (a << (shift + 4)) + c;

        __syncthreads();   // previous tile fully consumed
        #pragma unroll
        for (int i = 0; i < 16; ++i)
            tile[tid][i] = vb[base + ((unsigned)i << shift)];
        __syncthreads();

        #pragma unroll
        for (int s = 0; s < 8; ++s) {
            const int kb = wv * 32 + s * 4;
            v2f A, Bv;
            // A (16x4): lanes0-15 K={0,1}, lanes16-31 K={2,3}; A[m][k]=S[k][m]
            A.x  = tile[kb + 2 * h    ][m];
            A.y  = tile[kb + 2 * h + 1][m];
            // B (4x16): rows striped across lane halves per VGPR; B[k][n]=S[k][n]
            Bv.x = tile[kb + h        ][m];
            Bv.y = tile[kb + 2 + h    ][m];
            acc = __builtin_amdgcn_wmma_f32_16x16x4_f32(
                false, A, false, Bv, (short)0, acc, false, false);
        }
    }

    // out += sum_{M,N} T[M][N] * G[M][N]; C/D layout: VGPR v -> M=v+8h, N=m
    float partial = 0.f;
    #pragma unroll
    for (int v = 0; v < 8; ++v)
        partial = fmaf(Tl[(v + 8 * h) * 16 + m], acc[v], partial);
    #pragma unroll
    for (int o = 16; o; o >>= 1) partial += __shfl_xor(partial, o, 32);
    if (lane == 0) atomicAdd(&raw[b * 17 + q], partial);
}

// ---------------------------------------------------------------------------
// Kernel D: out[b,q] = raw[b,q] / max(||v_b||^2, 1e-24)
// ---------------------------------------------------------------------------
__global__ void finalize_kernel(const float* __restrict__ raw,
                                const float* __restrict__ sumsq,
                                float* __restrict__ out) {
    const int idx = blockIdx.x * 32 + threadIdx.x;   // grid 17 x 32 = 544
    if (idx < 544) {
        const int b = idx / 17;
        out[idx] = raw[idx] / fmaxf(sumsq[b], 1e-24f);
    }
}

extern "C" void kernel_launch(void* const* d_in, const int* in_sizes, int n_in,
                              void* d_out, int out_size, void* d_ws, size_t ws_size,
                              hipStream_t stream) {
    (void)in_sizes; (void)n_in; (void)out_size; (void)ws_size;
    const float* vec = (const float*)d_in[0];   // (32, 2^20) f32
    const float* w   = (const float*)d_in[1];   // (36,) f32
    float* out = (float*)d_out;                 // (32, 17) f32

    float* wsf   = (float*)d_ws;
    float* Tg    = wsf;         // 256 floats: T = Re(U' M U)
    float* sumsq = wsf + 256;   // 32 floats
    float* raw   = wsf + 288;   // 544 floats (atomic accumulators)

    hipMemsetAsync(raw, 0, 544 * sizeof(float), stream);
    build_temp_kernel<<<1, 256, 0, stream>>>(w, Tg);
    sumsq_kernel<<<32, 256, 0, stream>>>(vec, sumsq);
    gram_kernel<<<dim3(32, 17, 32), 256, 0, stream>>>(vec, Tg, raw);
    finalize_kernel<<<17, 32, 0, stream>>>(raw, sumsq, out);
}